// GAT_29334626632214
// MI455X (gfx1250) — compile-verified
//
#include <hip/hip_runtime.h>
#include <hip/hip_bf16.h>

typedef __attribute__((ext_vector_type(2))) float v2f;
typedef __attribute__((ext_vector_type(8))) float v8f;

#define NEG_SLOPE 0.2f
#define GAT_EPS 1e-16f

// ---- order-preserving float<->uint encoding for atomicMax on floats ----
__device__ __forceinline__ unsigned fenc(float f) {
  unsigned u = __float_as_uint(f);
  return (u & 0x80000000u) ? ~u : (u | 0x80000000u);
}
__device__ __forceinline__ float fdec(unsigned e) {
  unsigned u = (e & 0x80000000u) ? (e ^ 0x80000000u) : ~e;
  return __uint_as_float(u);
}
// fenc(-inf) = ~0xFF800000 = 0x007FFFFF
#define ENC_NEG_INF 0x007FFFFFu

__global__ void gat_fill_u32(unsigned* __restrict__ p, unsigned v, int n) {
  int i = blockIdx.x * blockDim.x + threadIdx.x;
  if (i < n) p[i] = v;
}

// ---------------------------------------------------------------------------
// C[N,Fout] = A[N,K] @ B[K,Fout], full fp32 via V_WMMA_F32_16X16X4_F32.
// One wave per 16x16 tile. Requires N%16==0, K%4==0, Fout%16==0. blockDim=(32,8).
// ---------------------------------------------------------------------------
__global__ void gat_gemm_wmma(const float* __restrict__ A, const float* __restrict__ B,
                              float* __restrict__ C, int N, int K, int Fout) {
  const int lane = threadIdx.x & 31;
  const int half = lane >> 4;   // 0: K lanes {0,1}/rows M..  1: K lanes {2,3}
  const int l16  = lane & 15;
  const int mt = blockIdx.x;
  const int nt = blockIdx.y * blockDim.y + threadIdx.y;
  if (nt * 16 >= Fout) return;  // whole-wave exit: EXEC stays all-ones for WMMA

  const int arow = mt * 16 + l16;      // A: lane = M row
  const int bcol = nt * 16 + l16;      // B/C: lane = N col
  const float* Ap = A + (size_t)arow * K;

  v8f acc = {};
  for (int k0 = 0; k0 < K; k0 += 4) {
    const int ka = k0 + half * 2;
    v2f a, b;
    a.x = Ap[ka];
    a.y = Ap[ka + 1];
    b.x = B[(size_t)ka * Fout + bcol];
    b.y = B[(size_t)(ka + 1) * Fout + bcol];
    acc = __builtin_amdgcn_wmma_f32_16x16x4_f32(
        /*neg_a=*/false, a, /*neg_b=*/false, b,
        /*c_mod=*/(short)0, acc, /*reuse_a=*/false, /*reuse_b=*/false);
  }
#pragma unroll
  for (int r = 0; r < 8; ++r) {
    C[(size_t)(mt * 16 + half * 8 + r) * Fout + bcol] = acc[r];
  }
}

// ---------------------------------------------------------------------------
// a_s[n,h] = dot(h[n,h,:], att_src[h,:]);  a_d likewise. One wave per (n,h).
// ---------------------------------------------------------------------------
__global__ void gat_attn_scores(const float* __restrict__ h,
                                const float* __restrict__ att_src,
                                const float* __restrict__ att_dst,
                                float* __restrict__ a_s, float* __restrict__ a_d,
                                int N, int H, int Cc) {
  const int wid  = blockIdx.x * (blockDim.x >> 5) + (threadIdx.x >> 5);
  const int lane = threadIdx.x & 31;
  if (wid >= N * H) return;
  const int hh = wid % H;
  const float* hp = h + (size_t)wid * Cc;  // h is [N,H,C] contiguous; wid == n*H+hh
  float s = 0.f, d = 0.f;
  for (int c = lane; c < Cc; c += 32) {
    float v = hp[c];
    s += v * att_src[hh * Cc + c];
    d += v * att_dst[hh * Cc + c];
  }
#pragma unroll
  for (int off = 16; off > 0; off >>= 1) {
    s += __shfl_xor(s, off, 32);
    d += __shfl_xor(d, off, 32);
  }
  if (lane == 0) { a_s[wid] = s; a_d[wid] = d; }
}

// ---------------------------------------------------------------------------
// Pass 1: segment max of leaky-relu logits into encoded-uint m[dst,h].
// One thread per (edge, head); edges e >= E are the implicit self-loops.
// ---------------------------------------------------------------------------
__global__ void gat_edge_max(const int* __restrict__ src, const int* __restrict__ dst,
                             int E, int N, int H,
                             const float* __restrict__ a_s, const float* __restrict__ a_d,
                             unsigned* __restrict__ m_enc) {
  int t = blockIdx.x * blockDim.x + threadIdx.x;
  int total = (E + N) * H;
  if (t >= total) return;
  int e  = t / H;
  int hh = t - e * H;
  int s = (e < E) ? src[e] : (e - E);
  int d = (e < E) ? dst[e] : (e - E);
  float logit = a_s[s * H + hh] + a_d[d * H + hh];
  logit = (logit > 0.f) ? logit : NEG_SLOPE * logit;
  atomicMax(&m_enc[d * H + hh], fenc(logit));
}

// ---------------------------------------------------------------------------
// Pass 2 (fused): el = exp(logit - m[dst]); denom[dst] += el;
//                 num[dst,h,:] += el * h[src,h,:].   One wave per edge.
// ---------------------------------------------------------------------------
__global__ void gat_edge_agg(const int* __restrict__ src, const int* __restrict__ dst,
                             int E, int N, int H, int Cc,
                             const float* __restrict__ a_s, const float* __restrict__ a_d,
                             const unsigned* __restrict__ m_enc,
                             const float* __restrict__ h,
                             float* __restrict__ denom, float* __restrict__ num) {
  const int wid  = blockIdx.x * (blockDim.x >> 5) + (threadIdx.x >> 5);
  const int lane = threadIdx.x & 31;
  if (wid >= E + N) return;
  const int s = (wid < E) ? src[wid] : (wid - E);
  const int d = (wid < E) ? dst[wid] : (wid - E);
  for (int hh = 0; hh < H; ++hh) {
    float logit = a_s[s * H + hh] + a_d[d * H + hh];
    logit = (logit > 0.f) ? logit : NEG_SLOPE * logit;
    const float el = __expf(logit - fdec(m_enc[d * H + hh]));
    if (lane == 0) atomicAdd(&denom[d * H + hh], el);
    const float* hp = h   + (size_t)s * H * Cc + hh * Cc;
    float*       np = num + (size_t)d * H * Cc + hh * Cc;
    for (int c = lane; c < Cc; c += 32) {
      atomicAdd(&np[c], el * hp[c]);
    }
  }
}

// ---------------------------------------------------------------------------
// out[n,h,c] = num[n,h,c] / (denom[n,h] + EPS) + bias[h*C+c]; optional ReLU.
// In-place (out may alias num) — purely elementwise.
// ---------------------------------------------------------------------------
__global__ void gat_finalize(const float* __restrict__ num, const float* __restrict__ denom,
                             const float* __restrict__ bias, float* __restrict__ out,
                             int N, int H, int Cc, int do_relu) {
  int t = blockIdx.x * blockDim.x + threadIdx.x;
  int total = N * H * Cc;
  if (t >= total) return;
  int hc = H * Cc;
  int n  = t / hc;
  int r  = t - n * hc;
  int hh = r / Cc;
  float v = num[t] / (denom[n * H + hh] + GAT_EPS) + bias[r];
  if (do_relu) v = fmaxf(v, 0.f);
  out[t] = v;
}

// ---------------------------------------------------------------------------

static inline unsigned cdiv(unsigned a, unsigned b) { return (a + b - 1) / b; }

extern "C" void kernel_launch(void* const* d_in, const int* in_sizes, int n_in,
                              void* d_out, int out_size, void* d_ws, size_t ws_size,
                              hipStream_t stream) {
  const float* x        = (const float*)d_in[0];
  const int*   eidx     = (const int*)d_in[1];
  const float* W1       = (const float*)d_in[2];
  const float* att_src1 = (const float*)d_in[3];
  const float* att_dst1 = (const float*)d_in[4];
  const float* bias1    = (const float*)d_in[5];
  const float* W2       = (const float*)d_in[6];
  const float* att_src2 = (const float*)d_in[7];
  const float* att_dst2 = (const float*)d_in[8];
  const float* bias2    = (const float*)d_in[9];

  const int IN = 128, HID = 64, H1 = 2, OUT = 32;
  const int N = in_sizes[0] / IN;      // 50000 (multiple of 16)
  const int E = in_sizes[1] / 2;       // 800000
  const int F1 = H1 * HID;             // 128
  const int* src = eidx;
  const int* dst = eidx + E;

  // ---- workspace layout (all fp32/u32, ~60 MB; L2-resident) ----
  float* h1   = (float*)d_ws;                       // N*128
  float* num1 = h1   + (size_t)N * F1;              // N*128 (also layer-2 input after finalize)
  float* as1  = num1 + (size_t)N * F1;              // N*H1
  float* ad1  = as1  + (size_t)N * H1;              // N*H1
  float* den1 = ad1  + (size_t)N * H1;              // N*H1
  unsigned* m1 = (unsigned*)(den1 + (size_t)N * H1);// N*H1
  float* h2   = (float*)(m1 + (size_t)N * H1);      // N*32
  float* as2  = h2   + (size_t)N * OUT;             // N
  float* ad2  = as2  + N;                           // N
  float* den2 = ad2  + N;                           // N
  unsigned* m2 = (unsigned*)(den2 + N);             // N

  const dim3 gemmBlk(32, 8, 1);
  const int  TPB = 256;
  const int  WPB = TPB / 32;

  // ================= layer 1 =================
  gat_fill_u32<<<cdiv(N * F1, TPB), TPB, 0, stream>>>((unsigned*)num1, 0u, N * F1);
  gat_fill_u32<<<cdiv(N * H1, TPB), TPB, 0, stream>>>((unsigned*)den1, 0u, N * H1);
  gat_fill_u32<<<cdiv(N * H1, TPB), TPB, 0, stream>>>(m1, ENC_NEG_INF, N * H1);

  gat_gemm_wmma<<<dim3(N / 16, cdiv(F1 / 16, 8)), gemmBlk, 0, stream>>>(x, W1, h1, N, IN, F1);
  gat_attn_scores<<<cdiv(N * H1, WPB), TPB, 0, stream>>>(h1, att_src1, att_dst1, as1, ad1, N, H1, HID);
  gat_edge_max<<<cdiv((E + N) * H1, TPB), TPB, 0, stream>>>(src, dst, E, N, H1, as1, ad1, m1);
  gat_edge_agg<<<cdiv(E + N, WPB), TPB, 0, stream>>>(src, dst, E, N, H1, HID,
                                                     as1, ad1, m1, h1, den1, num1);
  gat_finalize<<<cdiv(N * F1, TPB), TPB, 0, stream>>>(num1, den1, bias1, num1, N, H1, HID, 1);

  // ================= layer 2 =================
  float* out = (float*)d_out;
  gat_fill_u32<<<cdiv(N * OUT, TPB), TPB, 0, stream>>>((unsigned*)out, 0u, N * OUT);
  gat_fill_u32<<<cdiv(N, TPB), TPB, 0, stream>>>((unsigned*)den2, 0u, N);
  gat_fill_u32<<<cdiv(N, TPB), TPB, 0, stream>>>(m2, ENC_NEG_INF, N);

  gat_gemm_wmma<<<dim3(N / 16, cdiv(OUT / 16, 8)), gemmBlk, 0, stream>>>(num1, W2, h2, N, F1, OUT);
  gat_attn_scores<<<cdiv(N * 1, WPB), TPB, 0, stream>>>(h2, att_src2, att_dst2, as2, ad2, N, 1, OUT);
  gat_edge_max<<<cdiv((E + N) * 1, TPB), TPB, 0, stream>>>(src, dst, E, N, 1, as2, ad2, m2);
  gat_edge_agg<<<cdiv(E + N, WPB), TPB, 0, stream>>>(src, dst, E, N, 1, OUT,
                                                     as2, ad2, m2, h2, den2, out);
  gat_finalize<<<cdiv(N * OUT, TPB), TPB, 0, stream>>>(out, den2, bias2, out, N, 1, OUT, 0);
}